// BellmanFordLayer_40372692582737
// MI455X (gfx1250) — compile-verified
//
#include <hip/hip_runtime.h>
#include <hip/hip_bf16.h>
#include <math.h>

// Bellman-Ford, min-plus MV per step.  N = 2048.
#define NN 2048
#define MAIN_BLOCKS 64
#define TPB 256
#define COLS_PB 32              // columns per block (MAIN_BLOCKS * COLS_PB == NN)
#define SEGS 8                  // row segments per column (TPB / COLS_PB)
#define ROWS_PT (NN / SEGS)     // 256 rows per thread
#define HALF (ROWS_PT / 2)      // 128: two accumulator chains for ILP
#define TSTEPS 32               // output staging depth (steps per TDM flush)
#define REM ((NN - 1) % TSTEPS)         // 31 tail steps
#define FULLSTEPS ((NN - 1) - REM)      // 2016

typedef unsigned int u32x4 __attribute__((ext_vector_type(4)));
typedef int i32x4 __attribute__((ext_vector_type(4)));
typedef int i32x8 __attribute__((ext_vector_type(8)));

// ---------------------------------------------------------------------------
// Tensor Data Mover descriptors (CDNA5 ISA ch. 8, D# groups 0/1; groups 2/3
// zero => tensors up to 2D).
// ---------------------------------------------------------------------------
__device__ __forceinline__ u32x4 tdm_g0(unsigned lds_addr, const void* gaddr) {
  unsigned long long ga = (unsigned long long)(size_t)gaddr;
  u32x4 g;
  g.x = 1u;                                   // count=1 valid, user mode
  g.y = lds_addr;                             // LDS byte address
  g.z = (unsigned)ga;                         // global_addr[31:0]
  g.w = (unsigned)(ga >> 32) | (2u << 30);    // global_addr[56:32] | type=2
  return g;
}

// 2D tile: tile_d0 contiguous elements per row, tile_d1 rows, row stride
// stride0 (elements); data_size = 4 bytes (f32/i32). tile_d1==0 -> 1D.
__device__ __forceinline__ i32x8 tdm_g1(unsigned tensor_d0, unsigned tensor_d1,
                                        unsigned tile_d0, unsigned tile_d1,
                                        unsigned stride0) {
  i32x8 g;
  g.s0 = 0x00020000;                                          // data_size=4B
  g.s1 = (int)((tensor_d0 & 0xFFFFu) << 16);                  // dim0 lo16
  g.s2 = (int)((tensor_d0 >> 16) | ((tensor_d1 & 0xFFFFu) << 16));
  g.s3 = (int)((tensor_d1 >> 16) | ((tile_d0 & 0xFFFFu) << 16));
  g.s4 = (int)(tile_d1 & 0xFFFFu);                            // tile_dim1
  g.s5 = (int)stride0;                                        // dim0 stride
  g.s6 = 0;
  g.s7 = 0;
  return g;
}

__device__ __forceinline__ void tdm_load_to_lds(unsigned lds_addr,
                                                const void* gaddr, i32x8 g1) {
  u32x4 g0 = tdm_g0(lds_addr, gaddr);
  i32x4 z4 = {0, 0, 0, 0};
#if defined(__clang_major__) && __clang_major__ >= 23
  i32x8 z8 = {0, 0, 0, 0, 0, 0, 0, 0};
  __builtin_amdgcn_tensor_load_to_lds(g0, g1, z4, z4, z8, 16);   // scope DEV
#else
  __builtin_amdgcn_tensor_load_to_lds(g0, g1, z4, z4, 16);       // scope DEV
#endif
}

__device__ __forceinline__ void tdm_store_from_lds(unsigned lds_addr,
                                                   const void* gaddr,
                                                   i32x8 g1) {
  u32x4 g0 = tdm_g0(lds_addr, gaddr);
  i32x4 z4 = {0, 0, 0, 0};
#if defined(__clang_major__) && __clang_major__ >= 23
  i32x8 z8 = {0, 0, 0, 0, 0, 0, 0, 0};
  __builtin_amdgcn_tensor_store_from_lds(g0, g1, z4, z4, z8, 0);
#else
  __builtin_amdgcn_tensor_store_from_lds(g0, g1, z4, z4, 0);
#endif
}

// ---------------------------------------------------------------------------
// Async global->LDS staging (ASYNCcnt path), used for the negative-cycle pass.
// ---------------------------------------------------------------------------
__device__ __forceinline__ void stage_dvec_async(float* dprev_lds,
                                                 const float* src_global,
                                                 int tid) {
  unsigned lds_off = (unsigned)(size_t)(void*)(dprev_lds + tid * 8);
  unsigned goff    = (unsigned)(tid * 32);
  asm volatile(
      "global_load_async_to_lds_b128 %0, %1, %2 scope:SCOPE_DEV\n\t"
      "global_load_async_to_lds_b128 %0, %1, %2 offset:16 scope:SCOPE_DEV\n\t"
      "s_wait_asynccnt 0"
      :
      : "v"(lds_off), "v"(goff), "s"(src_global)
      : "memory");
}

// Device-wide barrier: monotonic counter, one arrival per block per epoch.
__device__ __forceinline__ void grid_sync(unsigned* counter, unsigned* target) {
  __syncthreads();
  if (threadIdx.x == 0) {
    __threadfence();
    atomicAdd(counter, 1u);
    *target += gridDim.x;
    while (__hip_atomic_load(counter, __ATOMIC_ACQUIRE,
                             __HIP_MEMORY_SCOPE_AGENT) < *target) {
      __builtin_amdgcn_s_sleep(2);
    }
  }
  __syncthreads();
}

__global__ void bf_init_kernel(const int* __restrict__ src_node,
                               float* __restrict__ dist_out,
                               int* __restrict__ pred_out,
                               float* __restrict__ flag_out,
                               float* __restrict__ dbuf0,
                               unsigned* __restrict__ counter) {
  int j = blockIdx.x * blockDim.x + threadIdx.x;
  if (j < NN) {
    int s = *src_node;
    float v = (j == s) ? 0.0f : INFINITY;
    dbuf0[j] = v;
    dist_out[(size_t)j * NN] = v;   // distances[:, 0] = d0
    pred_out[(size_t)j * NN] = 0;   // predecessors[:, 0] = 0
    if (j == 0) {
      *counter = 0u;
      *flag_out = 0.0f;
    }
  }
}

__global__ __launch_bounds__(TPB, 1) void bf_main_kernel(
    const float* __restrict__ adj,
    float* __restrict__ dist_out,   // [NN][NN], dist_out[j*NN + t]
    int* __restrict__ pred_out,     // [NN][NN]
    float* __restrict__ flag_out,
    float* __restrict__ dbuf0,
    float* __restrict__ dbuf1,
    unsigned* __restrict__ counter) {
  __shared__ __align__(16) float dprev[NN];              // 8 KB staged d_prev
  __shared__ __align__(16) float sdist[COLS_PB * TSTEPS]; // 4 KB out staging
  __shared__ __align__(16) int   spred[COLS_PB * TSTEPS]; // 4 KB out staging
  __shared__ float sval[TPB];
  __shared__ int   sidx[TPB];

  const int tid  = threadIdx.x;
  const int c    = tid & (COLS_PB - 1);       // local column (lane-major)
  const int rseg = tid >> 5;                  // row segment 0..7
  const int col  = blockIdx.x * COLS_PB + c;  // global column
  const int jblk = blockIdx.x * COLS_PB;      // first column of block
  const int i0   = rseg * ROWS_PT;
  const float* p0 = adj + (size_t)i0 * NN + col;           // rows [i0, i0+128)
  const float* p1 = p0 + (size_t)HALF * NN;                // rows [i0+128, +128)

  const unsigned dprev_a = (unsigned)(size_t)(void*)dprev;
  const unsigned sdist_a = (unsigned)(size_t)(void*)sdist;
  const unsigned spred_a = (unsigned)(size_t)(void*)spred;

  const i32x8 g1_vec   = tdm_g1(NN, 1, NN, 0, NN);                  // 1D 8 KB
  const i32x8 g1_flush = tdm_g1(TSTEPS, COLS_PB, TSTEPS, COLS_PB, NN);
  const i32x8 g1_tail  = tdm_g1(REM, COLS_PB, TSTEPS, COLS_PB, NN); // OOB-drop

  unsigned target = 0;

  for (int step = 1; step <= NN - 1; ++step) {
    const float* dsrc = (step & 1) ? dbuf0 : dbuf1;  // step 1 reads dbuf0
    float*       ddst = (step & 1) ? dbuf1 : dbuf0;

    if (tid == 0) {
      tdm_load_to_lds(dprev_a, dsrc, g1_vec);  // DMA 8 KB, TENSORcnt
      __builtin_amdgcn_s_wait_tensorcnt(0);    // also fences prior flushes
    }
    __syncthreads();

    // min-plus argmin over 256 rows, two 128-row chains for ILP;
    // strict '<' + ascending merge keeps JAX first-index tie-breaking.
    float b0 = INFINITY, b1 = INFINITY;
    int x0 = i0, x1 = i0 + HALF;
    for (int ib = 0; ib < HALF; ib += 64) {
      if (ib + 64 < HALF) {
        __builtin_prefetch(p0 + (size_t)(ib + 64) * NN, 0, 0);
        __builtin_prefetch(p1 + (size_t)(ib + 64) * NN, 0, 0);
      }
#pragma unroll 8
      for (int k = 0; k < 64; ++k) {
        const int i = i0 + ib + k;
        float c0 = dprev[i] + p0[(size_t)(ib + k) * NN];
        float c1 = dprev[i + HALF] + p1[(size_t)(ib + k) * NN];
        if (c0 < b0) { b0 = c0; x0 = i; }
        if (c1 < b1) { b1 = c1; x1 = i + HALF; }
      }
    }
    float best = b0; int bidx = x0;
    if (b1 < b0) { best = b1; bidx = x1; }

    sval[tid] = best;
    sidx[tid] = bidx;
    __syncthreads();

    if (rseg == 0) {  // wave 0: combine 8 segments in ascending row order
      float v = sval[c];
      int   x = sidx[c];
#pragma unroll
      for (int r = 1; r < SEGS; ++r) {
        float v2 = sval[r * COLS_PB + c];
        int   x2 = sidx[r * COLS_PB + c];
        if (v2 < v) { v = v2; x = x2; }
      }
      ddst[col] = v;                       // cross-block d_new
      const int slot = (step - 1) & (TSTEPS - 1);
      sdist[c * TSTEPS + slot] = v;        // LDS output staging
      spred[c * TSTEPS + slot] = x;
    }

    if (tid == 0 && (step & (TSTEPS - 1)) == 0) {
      asm volatile("s_wait_dscnt 0" ::: "memory");  // staging writes visible
      const int t0 = step - (TSTEPS - 1);
      tdm_store_from_lds(sdist_a, dist_out + (size_t)jblk * NN + t0, g1_flush);
      tdm_store_from_lds(spred_a, pred_out + (size_t)jblk * NN + t0, g1_flush);
    }

    grid_sync(counter, &target);
  }

  // Tail flush: steps 2017..2047 (31 slots); tile_d0=32, tensor_d0=31 ->
  // the 32nd element of each row is dropped by TDM OOB handling.
  if (tid == 0) {
    asm volatile("s_wait_dscnt 0" ::: "memory");
    tdm_store_from_lds(sdist_a, dist_out + (size_t)jblk * NN + (FULLSTEPS + 1),
                       g1_tail);
    tdm_store_from_lds(spred_a, pred_out + (size_t)jblk * NN + (FULLSTEPS + 1),
                       g1_tail);
  }

  // ---- negative-cycle check: any_i d_last[i] + A[i,j] < d_last[j] ----
  // d_last == dbuf1 (last step odd).  ASYNCcnt staging path.
  stage_dvec_async(dprev, dbuf1, tid);
  __syncthreads();

  float b0 = INFINITY, b1 = INFINITY;
  for (int ib = 0; ib < HALF; ib += 64) {
    if (ib + 64 < HALF) {
      __builtin_prefetch(p0 + (size_t)(ib + 64) * NN, 0, 0);
      __builtin_prefetch(p1 + (size_t)(ib + 64) * NN, 0, 0);
    }
#pragma unroll 8
    for (int k = 0; k < 64; ++k) {
      const int i = i0 + ib + k;
      b0 = fminf(b0, dprev[i] + p0[(size_t)(ib + k) * NN]);
      b1 = fminf(b1, dprev[i + HALF] + p1[(size_t)(ib + k) * NN]);
    }
  }
  sval[tid] = fminf(b0, b1);
  __syncthreads();
  if (rseg == 0) {
    float v = sval[c];
#pragma unroll
    for (int r = 1; r < SEGS; ++r) v = fminf(v, sval[r * COLS_PB + c]);
    if (v < dprev[col]) *flag_out = 1.0f;  // benign same-value race
  }
}

extern "C" void kernel_launch(void* const* d_in, const int* in_sizes, int n_in,
                              void* d_out, int out_size, void* d_ws,
                              size_t ws_size, hipStream_t stream) {
  const float* adj  = (const float*)d_in[0];
  const int*   srcn = (const int*)d_in[1];

  float* out  = (float*)d_out;
  float* dist = out;                                  // N*N f32
  int*   pred = (int*)(out + (size_t)NN * NN);        // N*N i32 (bit pattern)
  float* flag = out + 2 * (size_t)NN * NN;            // 1 scalar

  char*     ws      = (char*)d_ws;
  unsigned* counter = (unsigned*)ws;                  // barrier counter
  float*    dbuf0   = (float*)(ws + 128);             // d vectors, double-buf
  float*    dbuf1   = dbuf0 + NN;

  bf_init_kernel<<<NN / TPB, TPB, 0, stream>>>(srcn, dist, pred, flag, dbuf0,
                                               counter);
  bf_main_kernel<<<MAIN_BLOCKS, TPB, 0, stream>>>(adj, dist, pred, flag, dbuf0,
                                                  dbuf1, counter);
}